// SpectralConv2d_50113678410393
// MI455X (gfx1250) — compile-verified
//
#include <hip/hip_runtime.h>

// ---------------------------------------------------------------------------
// FNO SpectralConv2d for MI455X (gfx1250, wave32, WMMA + async-LDS copies).
// Partial-DFT-as-GEMM formulation; all GEMMs on V_WMMA_F32_16X16X4_F32.
// Global->LDS staging uses GLOBAL_LOAD_ASYNC_TO_LDS_B128 (ASYNCcnt).
// ---------------------------------------------------------------------------

typedef __attribute__((ext_vector_type(2))) float v2f;
typedef __attribute__((ext_vector_type(8))) float v8f;

namespace {
constexpr int kNx = 256, kNy = 256;
constexpr int kM  = 32;                     // retained modes per axis
constexpr int kCin = 64, kCout = 64, kB = 16;
constexpr int kNMode = kM * kM;             // 1024

// workspace layout (float offsets)
constexpr size_t OFF_BASIS  = 0;                              // [256][64] cos|sin fwd basis
constexpr size_t OFF_BASIST = OFF_BASIS  + 256 * 64;          // [64][256] weighted inv-y basis
constexpr size_t OFF_XFT    = OFF_BASIST + 64 * 256;          // [2][1024][cin=64][b=16]
constexpr size_t OFF_WT     = OFF_XFT + 2ull * kNMode * kCin * kB;   // [1024][2][64][64]
constexpr size_t OFF_OM     = OFF_WT  + (size_t)kNMode * 2 * kCin * kCout; // [b][o][2][1024]
}

__device__ __forceinline__ v8f wmma4(v2f a, v2f b, v8f c) {
  // D = A(16x4 f32) * B(4x16 f32) + C(16x16 f32)
  return __builtin_amdgcn_wmma_f32_16x16x4_f32(false, a, false, b, (short)0, c,
                                               false, false);
}

// Async global->LDS copy, 16B per lane, tracked on ASYNCcnt.
__device__ __forceinline__ void async_copy16(unsigned lds_off, const void* gaddr) {
  asm volatile("global_load_async_to_lds_b128 %0, %1, off"
               :: "v"(lds_off),
                  "v"((unsigned long long)(size_t)gaddr)
               : "memory");
}
__device__ __forceinline__ void wait_async0() {
  asm volatile("s_wait_asynccnt 0" ::: "memory");
}
// low 32 bits of a generic LDS pointer == LDS byte offset (aperture rule)
__device__ __forceinline__ unsigned lds_addr(const void* p) {
  return (unsigned)(size_t)p;
}

// ---------------------------------------------------------------------------
// Kernel 0: build DFT bases.
// basis[n][k]  : k<32 -> cos(2*pi*n*k/256), k>=32 -> sin(2*pi*n*(k-32)/256)
// basisT[k][y] : inverse-y basis with Hermitian doubling (c0=1, ck=2) and the
//                irfft2 ortho 1/256 folded in.  rows 0..31 cos, 32..63 sin.
// ---------------------------------------------------------------------------
__global__ void fno_init_basis(float* __restrict__ ws) {
  const int k = blockIdx.x;     // 0..63
  const int n = threadIdx.x;    // 0..255
  const float w0 = 6.2831853071795864769f / 256.0f;
  float* basis  = ws + OFF_BASIS;
  float* basisT = ws + OFF_BASIST;
  if (k < kM) {
    basis[n * 64 + k] = __cosf(w0 * (float)(n * k));
    const float wk = (k == 0 ? 1.0f : 2.0f) / 256.0f;
    basisT[k * 256 + n] = wk * __cosf(w0 * (float)(n * k));
  } else {
    const int kk = k - kM;
    basis[n * 64 + k] = __sinf(w0 * (float)(n * kk));
    const float wk = (kk == 0 ? 1.0f : 2.0f) / 256.0f;
    basisT[k * 256 + n] = wk * __sinf(w0 * (float)(n * kk));
  }
}

// ---------------------------------------------------------------------------
// Kernel 1: transpose weights [cin][cout][kx][ky][2] -> wT[mode][ri][cin][cout]
// so that the per-mode mixing GEMM reads B-fragments coalesced.
// ---------------------------------------------------------------------------
__global__ void fno_wtrans(const float* __restrict__ w, float* __restrict__ ws) {
  float* wT = ws + OFF_WT;
  const int mode = blockIdx.x;          // 0..1023
  const int o  = threadIdx.x & 63;
  const int ig = threadIdx.x >> 6;      // 0..3
  for (int i = ig; i < kCin; i += 4) {
    const size_t rd = ((size_t)(i * kCout + o) * kNMode + mode) * 2;
    wT[(size_t)mode * 8192 + 0 * 4096 + i * 64 + o] = w[rd + 0];
    wT[(size_t)mode * 8192 + 1 * 4096 + i * 64 + o] = w[rd + 1];
  }
}

// ---------------------------------------------------------------------------
// Kernel 2: forward partial DFT.  One block per (b,cin) image, 4 waves.
// Stage 1: T1[256][64] = X @ [cosY|sinY]        (GEMM 256x256x64)
//          A-slabs staged via async global->LDS copies (padded stride 260
//          floats -> conflict-free ds_load_b64 fragment reads).
// Stage 2: G1=Cx@T1, G2=Sx@T1 (GEMM 32x256x64 each), then
//          R =  (CxP - SxQ)/256,  I = -(SxP + CxQ)/256
// Output xft[ri][mode][cin][b].
// ---------------------------------------------------------------------------
#define XS_STRIDE 260   // 16 rows, padded: banks (4*m + k) % 64, conflict-free

__global__ void __launch_bounds__(128)
fno_fwd(const float* __restrict__ x, float* __restrict__ ws) {
  __shared__ float T1[256 * 64];                       // 64 KB
  __shared__ __align__(16) float Xs[4][16 * XS_STRIDE]; // ~65 KB, per-wave slab
  __shared__ float G[2][32 * 64];                      // 16 KB

  const float* basis = ws + OFF_BASIS;
  float* xft = ws + OFF_XFT;

  const int b  = blockIdx.x >> 6;
  const int ci = blockIdx.x & 63;
  const float* X = x + (size_t)(b * kCin + ci) * kNx * kNy;

  const int wave = threadIdx.x >> 5;
  const int lane = threadIdx.x & 31;
  const int mN = lane & 15;          // N (or M) index within fragment
  const int kb = (lane >> 4) * 2;    // K sub-offset: 0 or 2

  // ---- stage 1: T1 = X @ basis ----
  const unsigned xsBase = lds_addr(&Xs[wave][0]);
  for (int s = wave; s < 16; s += 4) {
    const int x0 = s * 16;
    // async-stage this wave's 16x256 slab (1 KB/row -> 2x512B chunks)
    const float* src = X + (size_t)x0 * 256;
    for (int r = 0; r < 16; ++r) {
      async_copy16(xsBase + r * (XS_STRIDE * 4) + lane * 16,
                   src + r * 256 + lane * 4);
      async_copy16(xsBase + r * (XS_STRIDE * 4) + 512 + lane * 16,
                   src + r * 256 + 128 + lane * 4);
    }
    wait_async0();

    v8f acc[4] = {};
    for (int k0 = 0; k0 < 256; k0 += 4) {
      v2f a;
      a.x = Xs[wave][mN * XS_STRIDE + k0 + kb + 0];
      a.y = Xs[wave][mN * XS_STRIDE + k0 + kb + 1];
      #pragma unroll
      for (int nt = 0; nt < 4; ++nt) {
        const int n = nt * 16 + mN;
        v2f bb;
        bb.x = basis[(k0 + kb + 0) * 64 + n];
        bb.y = basis[(k0 + kb + 1) * 64 + n];
        acc[nt] = wmma4(a, bb, acc[nt]);
      }
    }
    #pragma unroll
    for (int nt = 0; nt < 4; ++nt)
      #pragma unroll
      for (int j = 0; j < 8; ++j)
        T1[(x0 + j + 8 * (lane >> 4)) * 64 + nt * 16 + mN] = acc[nt][j];
  }
  __syncthreads();

  // ---- stage 2: G[g] = (g? Sx : Cx) @ T1 ----
  {
    const int g  = wave >> 1;            // 0 = cos, 1 = sin
    const int m0 = (wave & 1) * 16;      // kx tile
    v8f acc[4] = {};
    for (int k0 = 0; k0 < 256; k0 += 4) {
      v2f a;   // A[m=kx][k=x] = basis[x][g*32 + kx]
      a.x = basis[(k0 + kb + 0) * 64 + g * 32 + m0 + mN];
      a.y = basis[(k0 + kb + 1) * 64 + g * 32 + m0 + mN];
      #pragma unroll
      for (int nt = 0; nt < 4; ++nt) {
        const int n = nt * 16 + mN;
        v2f bb;
        bb.x = T1[(k0 + kb + 0) * 64 + n];
        bb.y = T1[(k0 + kb + 1) * 64 + n];
        acc[nt] = wmma4(a, bb, acc[nt]);
      }
    }
    #pragma unroll
    for (int nt = 0; nt < 4; ++nt)
      #pragma unroll
      for (int j = 0; j < 8; ++j)
        G[g][(m0 + j + 8 * (lane >> 4)) * 64 + nt * 16 + mN] = acc[nt][j];
  }
  __syncthreads();

  // ---- combine + scatter to xft[ri][mode][ci][b] ----
  const float sc = 1.0f / 256.0f;   // forward ortho 1/sqrt(Nx*Ny)
  for (int t = threadIdx.x; t < kNMode; t += 128) {
    const int kx = t >> 5, ky = t & 31;
    const float p1 = G[0][kx * 64 + ky];        // Cx@P
    const float q1 = G[0][kx * 64 + 32 + ky];   // Cx@Q
    const float p2 = G[1][kx * 64 + ky];        // Sx@P
    const float q2 = G[1][kx * 64 + 32 + ky];   // Sx@Q
    const float R =  (p1 - q2) * sc;
    const float I = -(p2 + q1) * sc;
    xft[(size_t)0 * kNMode * 1024 + (size_t)t * 1024 + ci * 16 + b] = R;
    xft[(size_t)1 * kNMode * 1024 + (size_t)t * 1024 + ci * 16 + b] = I;
  }
}

// ---------------------------------------------------------------------------
// Kernel 3: per-mode complex channel mix.  One block per mode, 4 waves.
// wave c computes one of {XR@WR, XI@WI, XR@WI, XI@WR} over 4 N-tiles.
// Rout = c0 - c1 ; Iout = c2 + c3.  Output OM[b][o][ri][mode].
// ---------------------------------------------------------------------------
__global__ void __launch_bounds__(128)
fno_mix(float* __restrict__ ws) {
  __shared__ float Gm[4][16 * 64];  // 16 KB

  const int mode = blockIdx.x;
  const float* xft = ws + OFF_XFT;
  const float* XR = xft + (size_t)mode * 1024;
  const float* XI = XR + (size_t)kNMode * 1024;
  const float* wT = ws + OFF_WT + (size_t)mode * 8192;
  const float* WR = wT;
  const float* WI = wT + 4096;
  float* OM = ws + OFF_OM;

  const int wave = threadIdx.x >> 5;
  const int lane = threadIdx.x & 31;
  const int mN = lane & 15;
  const int kb = (lane >> 4) * 2;

  const float* Amat = (wave == 0 || wave == 2) ? XR : XI;  // [i][b]
  const float* Bmat = (wave == 0 || wave == 3) ? WR : WI;  // [i][o]

  v8f acc[4] = {};
  for (int k0 = 0; k0 < kCin; k0 += 4) {
    v2f a;
    a.x = Amat[(k0 + kb + 0) * 16 + mN];
    a.y = Amat[(k0 + kb + 1) * 16 + mN];
    #pragma unroll
    for (int nt = 0; nt < 4; ++nt) {
      const int n = nt * 16 + mN;
      v2f bb;
      bb.x = Bmat[(k0 + kb + 0) * 64 + n];
      bb.y = Bmat[(k0 + kb + 1) * 64 + n];
      acc[nt] = wmma4(a, bb, acc[nt]);
    }
  }
  #pragma unroll
  for (int nt = 0; nt < 4; ++nt)
    #pragma unroll
    for (int j = 0; j < 8; ++j)
      Gm[wave][(j + 8 * (lane >> 4)) * 64 + nt * 16 + mN] = acc[nt][j];
  __syncthreads();

  for (int t = threadIdx.x; t < kB * kCout; t += 128) {
    const int bb = t >> 6, o = t & 63;
    const float R = Gm[0][bb * 64 + o] - Gm[1][bb * 64 + o];
    const float I = Gm[2][bb * 64 + o] + Gm[3][bb * 64 + o];
    OM[((size_t)(bb * kCout + o) * 2 + 0) * kNMode + mode] = R;
    OM[((size_t)(bb * kCout + o) * 2 + 1) * kNMode + mode] = I;
  }
}

// ---------------------------------------------------------------------------
// Kernel 4: inverse partial DFT.  One block per (b,cout) image, 4 waves.
// Modes staged to LDS via async global->LDS copies.  Per 16-row x-slab:
//  stage A: Ure = Cx@R - Sx@I ; Uim = Sx@R + Cx@I   (GEMM 16x32x32 x4 combos)
//  stage B: out_slab[16][256] = [Ure | -Uim] @ basisT[64][256]
// ---------------------------------------------------------------------------
__global__ void __launch_bounds__(128)
fno_inv(float* __restrict__ ws, float* __restrict__ out) {
  __shared__ __align__(16) float smR[32 * 32];   // 4 KB
  __shared__ __align__(16) float smI[32 * 32];   // 4 KB
  __shared__ float P[4][16 * 32];                // 8 KB partial combos
  __shared__ float U[16 * 64];                   // 4 KB  [Ure | -Uim]

  const float* basis  = ws + OFF_BASIS;
  const float* basisT = ws + OFF_BASIST;
  const float* OM = ws + OFF_OM + (size_t)blockIdx.x * 2 * kNMode;

  const int b = blockIdx.x >> 6;
  const int o = blockIdx.x & 63;
  float* dst = out + (size_t)(b * kCout + o) * kNx * kNy;

  const int wave = threadIdx.x >> 5;
  const int lane = threadIdx.x & 31;
  const int mN = lane & 15;
  const int kb = (lane >> 4) * 2;

  // async-stage the (R,I) mode planes (4 KB each, 16B chunks per lane)
  {
    const unsigned rb = lds_addr(&smR[0]);
    const unsigned ib = lds_addr(&smI[0]);
    for (int c = threadIdx.x; c < 256; c += 128) {
      async_copy16(rb + c * 16, OM + c * 4);
      async_copy16(ib + c * 16, OM + kNMode + c * 4);
    }
    wait_async0();
  }
  __syncthreads();

  for (int s = 0; s < 16; ++s) {
    const int x0 = s * 16;
    // stage A: wave = combo {Cx@R, Sx@I, Sx@R, Cx@I}
    const bool useCos = (wave == 0 || wave == 3);
    const float* Bsrc = (wave == 0 || wave == 2) ? smR : smI;
    v8f accA[2] = {};
    for (int k0 = 0; k0 < kM; k0 += 4) {
      v2f a;   // A[m=x][k=kx] = basis[x][(cos?0:32)+kx]
      a.x = basis[(x0 + mN) * 64 + (useCos ? 0 : 32) + k0 + kb + 0];
      a.y = basis[(x0 + mN) * 64 + (useCos ? 0 : 32) + k0 + kb + 1];
      #pragma unroll
      for (int nt = 0; nt < 2; ++nt) {
        const int n = nt * 16 + mN;
        v2f bb;
        bb.x = Bsrc[(k0 + kb + 0) * 32 + n];
        bb.y = Bsrc[(k0 + kb + 1) * 32 + n];
        accA[nt] = wmma4(a, bb, accA[nt]);
      }
    }
    #pragma unroll
    for (int nt = 0; nt < 2; ++nt)
      #pragma unroll
      for (int j = 0; j < 8; ++j)
        P[wave][(j + 8 * (lane >> 4)) * 32 + nt * 16 + mN] = accA[nt][j];
    __syncthreads();

    for (int t = threadIdx.x; t < 512; t += 128) {
      const int r = t >> 5, ky = t & 31;
      U[r * 64 + ky]      =  P[0][t] - P[1][t];   //  Ure
      U[r * 64 + 32 + ky] = -(P[2][t] + P[3][t]); // -Uim
    }
    __syncthreads();

    // stage B: out_slab = U @ basisT ; wave owns 4 y N-tiles
    v8f accB[4] = {};
    for (int k0 = 0; k0 < 64; k0 += 4) {
      v2f a;
      a.x = U[mN * 64 + k0 + kb + 0];
      a.y = U[mN * 64 + k0 + kb + 1];
      #pragma unroll
      for (int q = 0; q < 4; ++q) {
        const int n = (wave * 4 + q) * 16 + mN;
        v2f bb;
        bb.x = basisT[(k0 + kb + 0) * 256 + n];
        bb.y = basisT[(k0 + kb + 1) * 256 + n];
        accB[q] = wmma4(a, bb, accB[q]);
      }
    }
    #pragma unroll
    for (int q = 0; q < 4; ++q)
      #pragma unroll
      for (int j = 0; j < 8; ++j) {
        const int row = x0 + j + 8 * (lane >> 4);
        dst[row * 256 + (wave * 4 + q) * 16 + mN] = accB[q][j];
      }
    __syncthreads();  // P/U reused next slab
  }
}

// ---------------------------------------------------------------------------
extern "C" void kernel_launch(void* const* d_in, const int* in_sizes, int n_in,
                              void* d_out, int out_size, void* d_ws, size_t ws_size,
                              hipStream_t stream) {
  (void)in_sizes; (void)n_in; (void)out_size; (void)ws_size;
  const float* x = (const float*)d_in[0];
  const float* w = (const float*)d_in[1];
  float* out = (float*)d_out;
  float* ws  = (float*)d_ws;   // needs ~48.2 MB

  fno_init_basis<<<dim3(64), dim3(256), 0, stream>>>(ws);
  fno_wtrans    <<<dim3(kNMode), dim3(256), 0, stream>>>(w, ws);
  fno_fwd       <<<dim3(kB * kCin), dim3(128), 0, stream>>>(x, ws);
  fno_mix       <<<dim3(kNMode), dim3(128), 0, stream>>>(ws);
  fno_inv       <<<dim3(kB * kCout), dim3(128), 0, stream>>>(ws, out);
}